// MultiFreqEncoder2D_8581344657663
// MI455X (gfx1250) — compile-verified
//
#include <hip/hip_runtime.h>
#include <math.h>
#include <stdint.h>

#define DEVINL __device__ __forceinline__

// ---- CDNA5 async global->LDS path (guarded; fallback = plain LDS copy) ----
#if defined(__gfx1250__) && __has_builtin(__builtin_amdgcn_global_load_async_to_lds_b128)
#define HAVE_ASYNC 1
#else
#define HAVE_ASYNC 0
#endif

typedef int v4i_t __attribute__((vector_size(16)));
typedef __attribute__((address_space(1))) v4i_t gv4i_t;  // global v4i
typedef __attribute__((address_space(3))) v4i_t lv4i_t;  // LDS v4i

static constexpr int kC = 8;                               // feature channels
static constexpr int kR[8] = {4, 8, 16, 40, 92, 208, 460, 1024}; // grid res per scale

struct Args {
  const float* x;        // [N,3]
  const float* feat[8];  // each [6, 8, R, R]
  const float* feat2[8]; // each [6, 8, R, R]
  const float* scales;   // [8]
  float* out;            // [N, 387]
  int n;
};

DEVINL void stage_async(const float* g, float* l, int n4, int t, int nthr) {
#if HAVE_ASYNC
  for (int i = t; i < n4; i += nthr) {
    // (global src, lds dst, imm offset, imm cpol)
    __builtin_amdgcn_global_load_async_to_lds_b128(
        (gv4i_t*)(uintptr_t)(g + 4 * i),
        (lv4i_t*)(uint32_t)(uintptr_t)(l + 4 * i),
        0, 0);
  }
#else
  for (int i = t; i < n4; i += nthr) {
    reinterpret_cast<float4*>(l)[i] = reinterpret_cast<const float4*>(g)[i];
  }
#endif
}

DEVINL void wait_async_done() {
#if HAVE_ASYNC
#if __has_builtin(__builtin_amdgcn_s_wait_asynccnt)
  __builtin_amdgcn_s_wait_asynccnt(0);
#else
  asm volatile("s_wait_asynccnt 0" ::: "memory");
#endif
#endif
}

template <bool NT>
DEVINL float ldg(const float* p) {
  return NT ? __builtin_nontemporal_load(p) : *p;
}

// One scale: 6 batch channels x 8 feature channels of (1D sample * 2D sample + lat)
template <bool NT>
DEVINL void sample_scale(const float* f1, const float* f2, const int R,
                         const float sc, const float x0, const float x1,
                         const float x2, float* op, const int obase) {
  const float kHalfPi = 1.57079632679489661923f;  // rounds to f32 pi/2, as in ref
  float sn[3], cs[3];
  {
    float v;
    v = x0 * sc; sn[0] = sinf(v); cs[0] = sinf(v + kHalfPi);
    v = x1 * sc; sn[1] = sinf(v); cs[1] = sinf(v + kHalfPi);
    v = x2 * sc; sn[2] = sinf(v); cs[2] = sinf(v + kHalfPi);
  }
  const int plane = R * R;
  const int xc = (R >> 1) - 1;          // floor(0.5*(R-1)) for even R; wx == 0.5 exactly
  const float rm1 = (float)(R - 1);
  const float rmax = (float)(R - 2);
#pragma unroll
  for (int b = 0; b < 6; ++b) {
    const int ph = b / 3;
    const int ax = b - 3 * ph;
    const float latv = ph ? cs[ax] : sn[ax];
    const float gxv = ph ? cs[(ax + 1) % 3] : sn[(ax + 1) % 3];
    const float gyv = ph ? cs[(ax + 2) % 3] : sn[(ax + 2) % 3];

    // fs: 1D bilinear along y (x fixed, wx=0.5)
    float py = (latv + 1.f) * 0.5f * rm1;
    float y0f = fminf(fmaxf(floorf(py), 0.f), rmax);
    float wy = py - y0f;
    int yi = (int)y0f;

    // fs2: full 2D bilinear
    float px2 = (gxv + 1.f) * 0.5f * rm1;
    float x0f = fminf(fmaxf(floorf(px2), 0.f), rmax);
    float wx2 = px2 - x0f;
    int xi2 = (int)x0f;
    float py2 = (gyv + 1.f) * 0.5f * rm1;
    float y0f2 = fminf(fmaxf(floorf(py2), 0.f), rmax);
    float wy2 = py2 - y0f2;
    int yi2 = (int)y0f2;

    const float* r1 = f1 + (size_t)(b * kC) * (size_t)plane + (size_t)yi * R + xc;
    const float* r2 = f2 + (size_t)(b * kC) * (size_t)plane + (size_t)yi2 * R + xi2;
    float* o = op + obase + b;
    const float iwy = 1.f - wy, iwx2 = 1.f - wx2, iwy2 = 1.f - wy2;
#pragma unroll
    for (int c = 0; c < kC; ++c) {
      float a00 = r1[0], a01 = r1[1], a10 = r1[R], a11 = r1[R + 1];
      float b00 = ldg<NT>(r2), b01 = ldg<NT>(r2 + 1);
      float b10 = ldg<NT>(r2 + R), b11 = ldg<NT>(r2 + R + 1);
      float fsv = (a00 * 0.5f + a01 * 0.5f) * iwy + (a10 * 0.5f + a11 * 0.5f) * wy;
      float top = b00 * iwx2 + b01 * wx2;
      float bot = b10 * iwx2 + b11 * wx2;
      float f2v = top * iwy2 + bot * wy2;
      o[(size_t)c * 48] = fsv * f2v + latv;  // out[n, c, s, ph, ax]
      r1 += plane;
      r2 += plane;
    }
  }
}

__global__ __launch_bounds__(256) void MultiFreqEncoder2D_kernel(Args a) {
  // Stage the two smallest (fully shared) tables into LDS via async copy.
  __shared__ __align__(16) float sF0[6 * 8 * 4 * 4];
  __shared__ __align__(16) float sG0[6 * 8 * 4 * 4];
  __shared__ __align__(16) float sF1[6 * 8 * 8 * 8];
  __shared__ __align__(16) float sG1[6 * 8 * 8 * 8];

  const int t = threadIdx.x;
  const int nthr = blockDim.x;
  stage_async(a.feat[0], sF0, (6 * 8 * 4 * 4) / 4, t, nthr);
  stage_async(a.feat2[0], sG0, (6 * 8 * 4 * 4) / 4, t, nthr);
  stage_async(a.feat[1], sF1, (6 * 8 * 8 * 8) / 4, t, nthr);
  stage_async(a.feat2[1], sG1, (6 * 8 * 8 * 8) / 4, t, nthr);
  wait_async_done();
  __syncthreads();

  const int p = blockIdx.x * blockDim.x + t;
  if (p >= a.n) return;

  const float x0 = a.x[3 * (size_t)p + 0];
  const float x1 = a.x[3 * (size_t)p + 1];
  const float x2 = a.x[3 * (size_t)p + 2];
  float* op = a.out + (size_t)p * 387;
  op[0] = x0;
  op[1] = x1;
  op[2] = x2;

  // Scales 0,1 from LDS (ds_load), scales 2..6 from L2-resident global,
  // scale 7 features_2d with non-temporal loads (protect L2 residency of the rest).
  sample_scale<false>(sF0, sG0, 4, a.scales[0], x0, x1, x2, op, 3 + 0 * 6);
  sample_scale<false>(sF1, sG1, 8, a.scales[1], x0, x1, x2, op, 3 + 1 * 6);
#pragma unroll 1
  for (int i = 2; i < 7; ++i)
    sample_scale<false>(a.feat[i], a.feat2[i], kR[i], a.scales[i], x0, x1, x2,
                        op, 3 + i * 6);
  sample_scale<true>(a.feat[7], a.feat2[7], 1024, a.scales[7], x0, x1, x2, op,
                     3 + 7 * 6);
}

extern "C" void kernel_launch(void* const* d_in, const int* in_sizes, int n_in,
                              void* d_out, int out_size, void* d_ws,
                              size_t ws_size, hipStream_t stream) {
  (void)n_in; (void)out_size; (void)d_ws; (void)ws_size;
  Args a;
  a.x = (const float*)d_in[0];
  for (int i = 0; i < 8; ++i) a.feat[i] = (const float*)d_in[1 + i];
  for (int i = 0; i < 8; ++i) a.feat2[i] = (const float*)d_in[9 + i];
  a.scales = (const float*)d_in[17];
  a.out = (float*)d_out;
  a.n = in_sizes[0] / 3;  // x is [N,3]

  const int block = 256;
  const int grid = (a.n + block - 1) / block;
  MultiFreqEncoder2D_kernel<<<grid, block, 0, stream>>>(a);
}